// CenterHead_39505109188937
// MI455X (gfx1250) — compile-verified
//
#include <hip/hip_runtime.h>
#include <math.h>

// ---------------- problem constants (from reference) ----------------
#define B_    8
#define CAT_  10
#define H_    512
#define W_    512
#define HW_   (H_ * W_)        // 262144
#define K_    500
#define NSLICE (B_ * CAT_)     // 80
#define NBINS 2048
#define CHUNK 4096             // floats per TDM chunk (16 KB)
#define NBUF  4                // staging buffers -> up to 3 outstanding TDM ops
#define SPLIT 4                // segment-blocks per slice (320 blocks total)
#define SEG   (HW_ / SPLIT)    // 65536 floats per segment
#define NCHUNK_SEG (SEG / CHUNK)  // 16
#define CAND_CAP 2048          // per-slice candidate buffer (power of 2)
#define CAND2    8192          // stage-2 sort width (>= CAT_*K_ = 5000)

typedef unsigned int       u32;
typedef unsigned long long u64;

// ---------------- CDNA5 TDM availability ----------------
#if defined(__gfx1250__) && __has_builtin(__builtin_amdgcn_tensor_load_to_lds)
#define USE_TDM 1
#else
#define USE_TDM 0
#endif

typedef __attribute__((ext_vector_type(4))) unsigned int v4u;
typedef __attribute__((ext_vector_type(8))) int          v8i;
typedef __attribute__((ext_vector_type(4))) int          v4i;

// Order-preserving float->u32 key transform.
__device__ __forceinline__ u32 fkey(float f) {
    u32 b = __float_as_uint(f);
    return (b & 0x80000000u) ? ~b : (b | 0x80000000u);
}
__device__ __forceinline__ float unfkey(u32 ob) {
    u32 b = (ob & 0x80000000u) ? (ob & 0x7FFFFFFFu) : ~ob;
    return __uint_as_float(b);
}

__device__ __forceinline__ int score_bin(float v) {
    int bin = (int)(v * (float)NBINS);
    return bin < 0 ? 0 : (bin > NBINS - 1 ? NBINS - 1 : bin);
}

#if USE_TDM
// 1-D TDM copy of CHUNK contiguous f32 from global to LDS.
// amdgpu-toolchain (clang-23) 6-arg form: (v4u, v8i, v4i, v4i, v8i, i32 cpol).
__device__ __forceinline__ void tdm_load_chunk(const float* gsrc, float* ldsdst) {
    u64 ga = (u64)(const void*)gsrc;
    u32 la = (u32)(u64)(void*)ldsdst;   // low 32 bits of generic LDS addr = LDS offset
    v4u g0 = { 1u,                                              // count=1
               la,                                              // lds_addr
               (u32)ga,                                         // global_addr[31:0]
               (u32)((ga >> 32) & 0x1FFFFFFu) | (2u << 30) };   // addr[56:32] | type=2
    v8i g1 = { (int)(2u << 16),                    // data_size=4B, mask=0
               (int)((CHUNK & 0xFFFF) << 16),      // tensor_dim0[15:0]<<16
               (int)((CHUNK >> 16) | (1 << 16)),   // tensor_dim0[31:16] | tensor_dim1=1
               (int)((CHUNK & 0xFFFF) << 16),      // tile_dim0 = CHUNK
               0,                                  // tile_dim1/2 = 0
               (int)CHUNK,                         // tensor_dim0_stride[31:0]
               0, 0 };
    v4i gz4 = { 0, 0, 0, 0 };
    v8i gz8 = { 0, 0, 0, 0, 0, 0, 0, 0 };
    __builtin_amdgcn_tensor_load_to_lds(g0, g1, gz4, gz4, gz8, 0);
}
#endif

// Shared streaming prologue/step: deep TDM pipeline (3 outstanding).
#define TDM_PROLOGUE(srcp, stage_)                                              \
    if (tid == 0) {                                                             \
        _Pragma("unroll")                                                       \
        for (int p = 0; p < 3 && p < NCHUNK_SEG; ++p)                           \
            tdm_load_chunk((srcp) + (size_t)p * CHUNK, &(stage_)[p][0]);        \
    }

#define TDM_STEP(srcp, stage_, c)                                               \
    if (tid == 0) {                                                             \
        if ((c) + 3 < NCHUNK_SEG) {                                             \
            tdm_load_chunk((srcp) + (size_t)((c) + 3) * CHUNK,                  \
                           &(stage_)[((c) + 3) & (NBUF - 1)][0]);               \
            __builtin_amdgcn_s_wait_tensorcnt(3);                               \
        } else if ((c) == NCHUNK_SEG - 1) {                                     \
            __builtin_amdgcn_s_wait_tensorcnt(0);                               \
        } else if ((c) == NCHUNK_SEG - 2) {                                     \
            __builtin_amdgcn_s_wait_tensorcnt(1);                               \
        } else {                                                                \
            __builtin_amdgcn_s_wait_tensorcnt(2);                               \
        }                                                                       \
    }

// =====================================================================
// Kernel 0: zero global histogram + candidate counters.
// =====================================================================
__global__ void k_init(u32* __restrict__ g_hist, u32* __restrict__ g_cnt) {
    const int idx = blockIdx.x * blockDim.x + threadIdx.x;
    if (idx < NSLICE * NBINS) g_hist[idx] = 0;
    if (idx < NSLICE)         g_cnt[idx]  = 0;
}

// =====================================================================
// Kernel 1: segment histogram (SPLIT blocks per slice, 320 total).
// =====================================================================
__global__ __launch_bounds__(1024)
void k_hist(const float* __restrict__ heat, u32* __restrict__ g_hist) {
    __shared__ u32   hist[NBINS];            // 8 KB
    __shared__ float stage[NBUF][CHUNK];     // 64 KB
    const int tid   = threadIdx.x;
    const int slice = blockIdx.x / SPLIT;
    const int seg   = blockIdx.x % SPLIT;
    const float* src = heat + (size_t)slice * HW_ + (size_t)seg * SEG;

    for (int i = tid; i < NBINS; i += 1024) hist[i] = 0;
    __syncthreads();

#if USE_TDM
    TDM_PROLOGUE(src, stage)
#endif
    for (int c = 0; c < NCHUNK_SEG; ++c) {
        const int buf = c & (NBUF - 1);
#if USE_TDM
        TDM_STEP(src, stage, c)
#else
        {
            const float4* s4 = (const float4*)(src + (size_t)c * CHUNK);
            float4* d4 = (float4*)&stage[buf][0];
            for (int i = tid; i < CHUNK / 4; i += 1024) d4[i] = s4[i];
        }
#endif
        __syncthreads();
        #pragma unroll
        for (int r = 0; r < CHUNK / 1024; ++r)
            atomicAdd(&hist[score_bin(stage[buf][tid + r * 1024])], 1u);
        __syncthreads();
    }
    for (int i = tid; i < NBINS; i += 1024)
        atomicAdd(&g_hist[slice * NBINS + i], hist[i]);
}

// =====================================================================
// Kernel 2: threshold from global histogram + candidate compaction.
// =====================================================================
__global__ __launch_bounds__(1024)
void k_compact(const float* __restrict__ heat,
               const u32* __restrict__ g_hist,
               u64* __restrict__ g_cand,
               u32* __restrict__ g_cnt) {
    __shared__ float stage[NBUF][CHUNK];     // 64 KB
    __shared__ u32   csum[NBINS / 64];       // 32 coarse sums
    __shared__ int   s_thresh;
    const int tid   = threadIdx.x;
    const int slice = blockIdx.x / SPLIT;
    const int seg   = blockIdx.x % SPLIT;
    const float* src = heat + (size_t)slice * HW_ + (size_t)seg * SEG;
    const u32* hist = g_hist + slice * NBINS;

    // ---- threshold bin: suffix count >= K (redundant per block, cheap) ----
    if (tid < NBINS / 64) {
        u32 s = 0;
        #pragma unroll 8
        for (int i = 0; i < 64; ++i) s += hist[tid * 64 + i];
        csum[tid] = s;
    }
    __syncthreads();
    if (tid == 0) {
        u32 acc = 0;
        int cb = NBINS / 64 - 1;
        for (; cb >= 0; --cb) {
            if (acc + csum[cb] >= (u32)K_) break;
            acc += csum[cb];
        }
        int t = 0;
        if (cb >= 0) {
            int bin = cb * 64 + 63;
            for (; bin > cb * 64; --bin) {
                acc += hist[bin];
                if (acc >= (u32)K_) break;
            }
            if (acc < (u32)K_) bin = cb * 64;
            t = bin;
        }
        s_thresh = t;
    }
    __syncthreads();
    const int thresh = s_thresh;

    // ---- stream segment, append candidates >= threshold ----
#if USE_TDM
    TDM_PROLOGUE(src, stage)
#endif
    for (int c = 0; c < NCHUNK_SEG; ++c) {
        const int buf = c & (NBUF - 1);
#if USE_TDM
        TDM_STEP(src, stage, c)
#else
        {
            const float4* s4 = (const float4*)(src + (size_t)c * CHUNK);
            float4* d4 = (float4*)&stage[buf][0];
            for (int i = tid; i < CHUNK / 4; i += 1024) d4[i] = s4[i];
        }
#endif
        __syncthreads();
        #pragma unroll
        for (int r = 0; r < CHUNK / 1024; ++r) {
            const int li = tid + r * 1024;
            float v = stage[buf][li];
            if (score_bin(v) >= thresh) {
                u32 p = atomicAdd(&g_cnt[slice], 1u);
                if (p < CAND_CAP) {
                    u32 idx = (u32)(seg * SEG + c * CHUNK + li);
                    // high: orderable score bits; low: ~idx so ties pick lowest idx
                    g_cand[(size_t)slice * CAND_CAP + p] =
                        ((u64)fkey(v) << 32) | (u64)(0xFFFFFFFFu - idx);
                }
            }
        }
        __syncthreads();
    }
}

// =====================================================================
// Kernel 3: per-slice bitonic sort of candidates -> top-K.
// =====================================================================
__global__ __launch_bounds__(1024)
void k_sortk(const u64* __restrict__ g_cand,
             const u32* __restrict__ g_cnt,
             float* __restrict__ ws_scores,
             int*   __restrict__ ws_inds) {
    __shared__ u64 arr[CAND_CAP];            // 16 KB
    const int tid   = threadIdx.x;
    const int slice = blockIdx.x;
    u32 n = g_cnt[slice]; if (n > CAND_CAP) n = CAND_CAP;

    for (int i = tid; i < CAND_CAP; i += 1024)
        arr[i] = (i < (int)n) ? g_cand[(size_t)slice * CAND_CAP + i] : 0ull;
    __syncthreads();

    for (int k = 2; k <= CAND_CAP; k <<= 1) {
        for (int j = k >> 1; j > 0; j >>= 1) {
            for (int i = tid; i < CAND_CAP; i += 1024) {
                const int ixj = i ^ j;
                if (ixj > i) {
                    u64 a = arr[i], b2 = arr[ixj];
                    const bool dir = (i & k) != 0;     // overall descending
                    if ((a > b2) == dir) { arr[i] = b2; arr[ixj] = a; }
                }
            }
            __syncthreads();
        }
    }

    if (tid < K_) {
        u64 key = arr[tid];
        u32 ob  = (u32)(key >> 32);
        u32 idx = 0xFFFFFFFFu - (u32)key;
        ws_scores[slice * K_ + tid] = unfkey(ob);
        ws_inds  [slice * K_ + tid] = (int)idx;
    }
}

// =====================================================================
// Kernel 4: per-batch merge of CAT_*K_ candidates + gather + box decode.
// =====================================================================
__global__ __launch_bounds__(1024)
void k_merge_decode(const float* __restrict__ ws_scores,
                    const int*   __restrict__ ws_inds,
                    const float* __restrict__ rot_sine,
                    const float* __restrict__ rot_cosine,
                    const float* __restrict__ hei,
                    const float* __restrict__ dim3c,
                    const float* __restrict__ vel,
                    const float* __restrict__ reg,
                    float* __restrict__ out) {
    __shared__ u64 arr[CAND2];               // 64 KB
    const int tid = threadIdx.x;
    const int b   = blockIdx.x;
    const int NC  = CAT_ * K_;               // 5000

    for (int i = tid; i < CAND2; i += 1024) {
        u64 key = 0ull;
        if (i < NC) {
            float s = ws_scores[b * NC + i];
            key = ((u64)fkey(s) << 32) | (u64)(0xFFFFu - (u32)i);  // low idx wins ties
        }
        arr[i] = key;
    }
    __syncthreads();

    for (int k = 2; k <= CAND2; k <<= 1) {
        for (int j = k >> 1; j > 0; j >>= 1) {
            for (int i = tid; i < CAND2; i += 1024) {
                const int ixj = i ^ j;
                if (ixj > i) {
                    u64 a = arr[i], b2 = arr[ixj];
                    const bool dir = (i & k) != 0;
                    if ((a > b2) == dir) { arr[i] = b2; arr[ixj] = a; }
                }
            }
            __syncthreads();
        }
    }

    if (tid < K_) {
        u64 key  = arr[tid];
        u32 ob   = (u32)(key >> 32);
        int c    = (int)(0xFFFFu - (u32)(key & 0xFFFFFFFFull));
        float sc = unfkey(ob);
        int cls  = c / K_;
        int ind  = ws_inds[b * NC + c];

        float ysf = (float)(ind / W_);
        float xsf = (float)(ind % W_);

        const size_t base1 = (size_t)b * HW_ + (size_t)ind;
        float r0 = reg[(size_t)(b * 2 + 0) * HW_ + ind];
        float r1 = reg[(size_t)(b * 2 + 1) * HW_ + ind];
        float rs = rot_sine[base1];
        float rc = rot_cosine[base1];
        float hz = hei[base1];
        float d0 = dim3c[(size_t)(b * 3 + 0) * HW_ + ind];
        float d1 = dim3c[(size_t)(b * 3 + 1) * HW_ + ind];
        float d2 = dim3c[(size_t)(b * 3 + 2) * HW_ + ind];
        float v0 = vel[(size_t)(b * 2 + 0) * HW_ + ind];
        float v1 = vel[(size_t)(b * 2 + 1) * HW_ + ind];

        float x   = (xsf + r0) * 0.8f - 51.2f;   // OUT_SIZE_FACTOR*VOXEL + PC
        float y   = (ysf + r1) * 0.8f - 51.2f;
        float rot = atan2f(rs, rc);

        float* bx = out + ((size_t)b * K_ + tid) * 9;
        bx[0] = x;  bx[1] = y;  bx[2] = hz;
        bx[3] = d0; bx[4] = d1; bx[5] = d2;
        bx[6] = rot; bx[7] = v0; bx[8] = v1;

        const int OB = B_ * K_ * 9;              // 36000
        out[OB + b * K_ + tid]               = sc;
        out[OB + B_ * K_ + b * K_ + tid]     = (float)cls;
        const bool inside = (x >= -61.2f) && (x <= 61.2f) &&
                            (y >= -61.2f) && (y <= 61.2f) &&
                            (hz >= -10.0f) && (hz <= 10.0f);
        out[OB + 2 * B_ * K_ + b * K_ + tid] = (inside && (sc > 0.1f)) ? 1.0f : 0.0f;
    }
}

// =====================================================================
extern "C" void kernel_launch(void* const* d_in, const int* in_sizes, int n_in,
                              void* d_out, int out_size, void* d_ws, size_t ws_size,
                              hipStream_t stream) {
    const float* heat       = (const float*)d_in[0];
    const float* rot_sine   = (const float*)d_in[1];
    const float* rot_cosine = (const float*)d_in[2];
    const float* hei        = (const float*)d_in[3];
    const float* dim3c      = (const float*)d_in[4];
    const float* vel        = (const float*)d_in[5];
    const float* reg        = (const float*)d_in[6];

    // ---- workspace layout ----
    char* ws = (char*)d_ws;
    u32* g_hist      = (u32*)ws;                                   // 80*2048 u32
    u32* g_cnt       = (u32*)(ws + (size_t)NSLICE * NBINS * 4);    // 80 u32
    u64* g_cand      = (u64*)(ws + (size_t)NSLICE * NBINS * 4 + 320);  // 8B aligned
    float* ws_scores = (float*)((char*)g_cand + (size_t)NSLICE * CAND_CAP * 8);
    int*   ws_inds   = (int*)((char*)ws_scores + (size_t)NSLICE * K_ * 4);

    const int initN = NSLICE * NBINS;
    k_init<<<(initN + 255) / 256, 256, 0, stream>>>(g_hist, g_cnt);
    k_hist<<<NSLICE * SPLIT, 1024, 0, stream>>>(heat, g_hist);
    k_compact<<<NSLICE * SPLIT, 1024, 0, stream>>>(heat, g_hist, g_cand, g_cnt);
    k_sortk<<<NSLICE, 1024, 0, stream>>>(g_cand, g_cnt, ws_scores, ws_inds);
    k_merge_decode<<<B_, 1024, 0, stream>>>(ws_scores, ws_inds,
                                            rot_sine, rot_cosine, hei,
                                            dim3c, vel, reg, (float*)d_out);
}